// LaplaceLoss_35570919145577
// MI455X (gfx1250) — compile-verified
//
#include <hip/hip_runtime.h>
#include <hip/hip_bf16.h>
#include <math.h>

#define NSEG 1024
#define NPIX (512 * 512)
#define NB 4
#define NC 21

typedef float v2f __attribute__((ext_vector_type(2)));
typedef float v8f __attribute__((ext_vector_type(8)));

// ---------------------------------------------------------------------------
// Kernel 0: zero the accumulator region of the workspace (seg_sum, counts,
// m, w_ss, quad). Must run every launch: harness poisons ws once only.
// ---------------------------------------------------------------------------
__global__ void zero_ws_kernel(float* ws, int n) {
    int i = blockIdx.x * blockDim.x + threadIdx.x;
    if (i < n) ws[i] = 0.0f;
}

// ---------------------------------------------------------------------------
// Kernel 1: channel-sum + segment scatter. Each block handles a contiguous
// chunk of pixels of one batch; bins accumulated in LDS (ds_add_f32), then
// flushed with global fp32 atomics.  Streams 88 MB of pred coalesced.
// ---------------------------------------------------------------------------
__global__ void seg_kernel(const float* __restrict__ pred,
                           const int* __restrict__ ids,
                           float* __restrict__ seg_sum,
                           float* __restrict__ counts) {
    __shared__ float lsum[NSEG];
    __shared__ float lcnt[NSEG];
    const int b = blockIdx.y;
    const int t = threadIdx.x;             // 256 threads
    for (int i = t; i < NSEG; i += 256) { lsum[i] = 0.0f; lcnt[i] = 0.0f; }
    __syncthreads();

    const int chunk = NPIX / gridDim.x;    // gridDim.x = 64 -> 4096 px/block
    const int p0 = blockIdx.x * chunk;
    const float* pb = pred + (size_t)b * NC * NPIX;
    const int*   ib = ids  + (size_t)b * NPIX;

    for (int p = p0 + t; p < p0 + chunk; p += 256) {
        float s = 0.0f;
#pragma unroll
        for (int c = 0; c < NC; ++c)
            s += pb[(size_t)c * NPIX + p];
        const int id = ib[p];
        atomicAdd(&lsum[id], s);           // ds_add_f32
        atomicAdd(&lcnt[id], 1.0f);
    }
    __syncthreads();

    for (int i = t; i < NSEG; i += 256) {
        if (lcnt[i] != 0.0f) {
            atomicAdd(&seg_sum[b * NSEG + i], lsum[i]);  // global_atomic_add_f32
            atomicAdd(&counts [b * NSEG + i], lcnt[i]);
        }
    }
}

// ---------------------------------------------------------------------------
// Kernel 2: Frobenius sum-of-squares of Laplace_W per batch (16 MB stream).
// ---------------------------------------------------------------------------
__global__ void wss_kernel(const float* __restrict__ W, float* __restrict__ w_ss) {
    __shared__ float red[256];
    const int b = blockIdx.y;
    const float* Wb = W + (size_t)b * NSEG * NSEG;
    float s = 0.0f;
    for (int i = blockIdx.x * 256 + threadIdx.x; i < NSEG * NSEG;
         i += gridDim.x * 256) {
        const float v = Wb[i];
        s += v * v;
    }
    red[threadIdx.x] = s;
    __syncthreads();
    for (int off = 128; off > 0; off >>= 1) {
        if (threadIdx.x < off) red[threadIdx.x] += red[threadIdx.x + off];
        __syncthreads();
    }
    if (threadIdx.x == 0) atomicAdd(&w_ss[b], red[0]);
}

// ---------------------------------------------------------------------------
// Kernel 3: segment mean m = seg_sum / counts (counts>0 else 0).
// ---------------------------------------------------------------------------
__global__ void mean_kernel(const float* __restrict__ seg_sum,
                            const float* __restrict__ counts,
                            float* __restrict__ m) {
    const int i = blockIdx.x * blockDim.x + threadIdx.x;
    if (i < NB * NSEG) {
        const float c = counts[i];
        m[i] = (c > 0.0f) ? seg_sum[i] / c : 0.0f;
    }
}

// ---------------------------------------------------------------------------
// Kernel 4: quad[b] = m^T L m via V_WMMA_F32_16X16X4_F32.
// One wave per 16-row tile of L.  A = L[16 rows x 4 cols], B = m slice
// broadcast to all 16 columns, so D[i,j] = (L m)[row i] for every j.
//
// ISA VGPR layouts (32-bit, 16x16x4):
//   A (16x4):  lanes 0-15 -> K={0,1} in vgpr{0,1}; lanes 16-31 -> K={2,3}
//   B (4x16):  mirror of A with N = lane%16 (we broadcast over N)
//   C/D:       vgpr v, lane 0  -> (M=v,   N=0)
//              vgpr v, lane 16 -> (M=v+8, N=0)
// ---------------------------------------------------------------------------
__global__ void quad_kernel(const float* __restrict__ L,
                            const float* __restrict__ m,
                            float* __restrict__ quad) {
    __shared__ float sm[NSEG];
    const int b    = blockIdx.y;
    const int rt   = blockIdx.x;            // row tile 0..63
    const int lane = threadIdx.x;           // wave32, EXEC all ones

    const float* Lb = L + (size_t)b * NSEG * NSEG;
    const float* mb = m + b * NSEG;

    // stage m in LDS (uniform across the wave's K loop)
    for (int i = lane; i < NSEG; i += 32) sm[i] = mb[i];
    __syncthreads();

    const int row  = lane & 15;
    const int grow = rt * 16 + row;
    const int ksel = (lane >> 4) * 2;       // 0 for lanes 0-15, 2 for 16-31
    const float* rowp = Lb + (size_t)grow * NSEG;

    v8f acc = {};
    for (int k0 = 0; k0 < NSEG; k0 += 4) {
        v2f a, bv;
        a.x  = rowp[k0 + ksel];
        a.y  = rowp[k0 + ksel + 1];
        bv.x = sm[k0 + ksel];
        bv.y = sm[k0 + ksel + 1];
        acc = __builtin_amdgcn_wmma_f32_16x16x4_f32(
            /*neg_a=*/false, a, /*neg_b=*/false, bv,
            /*c_mod=*/(short)0, acc, /*reuse_a=*/false, /*reuse_b=*/false);
    }

    // column N=0 lives in lane 0 (rows 0..7) and lane 16 (rows 8..15)
    if (lane == 0 || lane == 16) {
        const int base = (lane == 0) ? 0 : 8;
        float s = 0.0f;
#pragma unroll
        for (int v = 0; v < 8; ++v)
            s += acc[v] * sm[rt * 16 + base + v];
        atomicAdd(&quad[b], s);
    }
}

// ---------------------------------------------------------------------------
// Kernel 5: loss = sum_b (2/sqrt(w_ss[b])) * 21 * quad[b]
// ---------------------------------------------------------------------------
__global__ void finalize_kernel(const float* __restrict__ w_ss,
                                const float* __restrict__ quad,
                                float* __restrict__ out) {
    if (threadIdx.x == 0 && blockIdx.x == 0) {
        float loss = 0.0f;
        for (int b = 0; b < NB; ++b)
            loss += (2.0f / sqrtf(w_ss[b])) * ((float)NC * quad[b]);
        out[0] = loss;
    }
}

// ---------------------------------------------------------------------------
// Workspace layout (floats):
//   [0      .. 4096)   seg_sum
//   [4096   .. 8192)   counts
//   [8192   .. 12288)  m
//   [12288  .. 12292)  w_ss
//   [12292  .. 12296)  quad
// ---------------------------------------------------------------------------
extern "C" void kernel_launch(void* const* d_in, const int* in_sizes, int n_in,
                              void* d_out, int out_size, void* d_ws, size_t ws_size,
                              hipStream_t stream) {
    const float* pred = (const float*)d_in[0];   // [4,21,512,512]
    const float* W    = (const float*)d_in[1];   // [4,1024,1024]
    const float* L    = (const float*)d_in[2];   // [4,1024,1024]
    const int*   ids  = (const int*)d_in[3];     // [4,512,512]

    float* ws      = (float*)d_ws;
    float* seg_sum = ws;
    float* counts  = ws + NB * NSEG;
    float* m       = ws + 2 * NB * NSEG;
    float* w_ss    = ws + 3 * NB * NSEG;
    float* quad    = w_ss + NB;

    const int nzero = 3 * NB * NSEG + 2 * NB;    // zero everything incl. m
    zero_ws_kernel<<<(nzero + 255) / 256, 256, 0, stream>>>(ws, nzero);

    seg_kernel<<<dim3(64, NB), 256, 0, stream>>>(pred, ids, seg_sum, counts);
    wss_kernel<<<dim3(64, NB), 256, 0, stream>>>(W, w_ss);

    mean_kernel<<<(NB * NSEG + 255) / 256, 256, 0, stream>>>(seg_sum, counts, m);

    quad_kernel<<<dim3(NSEG / 16, NB), 32, 0, stream>>>(L, m, quad);

    finalize_kernel<<<1, 32, 0, stream>>>(w_ss, quad, (float*)d_out);
}